// AttentivePool_28355374088835
// MI455X (gfx1250) — compile-verified
//
#include <hip/hip_runtime.h>

#define Bc 4
#define Cc 64
#define Nc 16384
#define Kc 16
#define Oc 128

typedef __attribute__((ext_vector_type(2))) float v2f;
typedef __attribute__((ext_vector_type(8))) float v8f;

// D(16x16,f32) = A(16x4,f32) * B(4x16,f32) + C   — full fp32 precision path.
__device__ __forceinline__ v8f wmma4(v2f a, v2f b, v8f c) {
  return __builtin_amdgcn_wmma_f32_16x16x4_f32(false, a, false, b, (short)0, c,
                                               false, false);
}

// ---------------------------------------------------------------------------
// Kernel 1: q/k/v = relu(BN(W @ feature)), written transposed as [B, N, C]
// so that the attention kernel's neighbor gathers are 256B-contiguous rows.
// Block = 128 threads (4 waves). Each wave owns one 16-row c_out tile and
// accumulates all three matrices (48 f32 WMMAs per wave).
// ---------------------------------------------------------------------------
__device__ __forceinline__ void spill_bn_relu_store(
    v8f acc, float* Os, int t, int w, int lrow, int hi,
    const float* __restrict__ g, const float* __restrict__ bb,
    const float* __restrict__ m, const float* __restrict__ vv,
    float* __restrict__ dst, int b, int n0) {
  __syncthreads();  // previous phase's LDS reads done
  float* ap = (float*)&acc;
#pragma unroll
  for (int r = 0; r < 8; ++r)
    Os[lrow * 65 + 16 * w + r + 8 * hi] = ap[r];  // D row = 16w + r + 8*hi
  __syncthreads();
#pragma unroll
  for (int i = 0; i < 8; ++i) {
    int lin = t + 128 * i;          // 0..1023 over [16 n][64 c]
    int n = lin >> 6, c = lin & 63;
    float sc = g[c] * __frsqrt_rn(vv[c] + 1e-5f);
    float y = (Os[n * 65 + c] - m[c]) * sc + bb[c];
    dst[((size_t)b * Nc + n0 + n) * Cc + c] = fmaxf(y, 0.0f);  // coalesced 256B rows
  }
}

__global__ __launch_bounds__(128) void qkv_kernel(
    const float* __restrict__ feature,
    const float* __restrict__ Wq_w, const float* __restrict__ Wq_g,
    const float* __restrict__ Wq_b, const float* __restrict__ Wq_m,
    const float* __restrict__ Wq_v,
    const float* __restrict__ Wk_w, const float* __restrict__ Wk_g,
    const float* __restrict__ Wk_b, const float* __restrict__ Wk_m,
    const float* __restrict__ Wk_v,
    const float* __restrict__ Wv_w, const float* __restrict__ Wv_g,
    const float* __restrict__ Wv_b, const float* __restrict__ Wv_m,
    const float* __restrict__ Wv_v,
    float* __restrict__ qT, float* __restrict__ kT, float* __restrict__ vT) {
  const int n0 = blockIdx.x * 16;
  const int b  = blockIdx.y;
  const int t  = threadIdx.x;

  __shared__ float Xs[Cc * 16];   // feature tile [c][n]
  __shared__ float Os[16 * 65];   // output staging [n][c] (pad 65)

  // Cooperative coalesced load of the 64x16 feature tile.
#pragma unroll
  for (int i = 0; i < 8; ++i) {
    int lin = t + 128 * i;
    int c = lin >> 4, j = lin & 15;
    Xs[lin] = feature[((size_t)b * Cc + c) * Nc + n0 + j];
  }
  __syncthreads();

  const int w = t >> 5, lane = t & 31, lrow = lane & 15, hi = lane >> 4;
  v8f aq = {0, 0, 0, 0, 0, 0, 0, 0};
  v8f ak = {0, 0, 0, 0, 0, 0, 0, 0};
  v8f av = {0, 0, 0, 0, 0, 0, 0, 0};
  const float* Aqp = Wq_w + (16 * w + lrow) * Cc;
  const float* Akp = Wk_w + (16 * w + lrow) * Cc;
  const float* Avp = Wv_w + (16 * w + lrow) * Cc;

#pragma unroll
  for (int kk = 0; kk < 16; ++kk) {        // K = 64 contraction, 4 per WMMA
    int k0 = 4 * kk + 2 * hi;
    v2f bm;
    bm.x = Xs[k0 * 16 + lrow];
    bm.y = Xs[(k0 + 1) * 16 + lrow];
    v2f a;
    a.x = Aqp[k0]; a.y = Aqp[k0 + 1]; aq = wmma4(a, bm, aq);
    a.x = Akp[k0]; a.y = Akp[k0 + 1]; ak = wmma4(a, bm, ak);
    a.x = Avp[k0]; a.y = Avp[k0 + 1]; av = wmma4(a, bm, av);
  }

  spill_bn_relu_store(aq, Os, t, w, lrow, hi, Wq_g, Wq_b, Wq_m, Wq_v, qT, b, n0);
  spill_bn_relu_store(ak, Os, t, w, lrow, hi, Wk_g, Wk_b, Wk_m, Wk_v, kT, b, n0);
  spill_bn_relu_store(av, Os, t, w, lrow, hi, Wv_g, Wv_b, Wv_m, Wv_v, vT, b, n0);
}

// ---------------------------------------------------------------------------
// Kernel 2: gather + attention + softmax(K=16, thread-local) + fused final
// 128x64 projection (WMMA) + BN(eps=1e-6) + LeakyReLU.
// Block = 256 threads (8 waves) handles 16 points.
// ---------------------------------------------------------------------------
__global__ __launch_bounds__(256) void attn_kernel(
    const float* __restrict__ x_info, const int* __restrict__ nbr,
    const float* __restrict__ qT, const float* __restrict__ kT,
    const float* __restrict__ vT,
    const float* __restrict__ m2_w, const float* __restrict__ m2_g,
    const float* __restrict__ m2_b, const float* __restrict__ m2_m,
    const float* __restrict__ m2_v,
    float* __restrict__ out) {
  const int n0 = blockIdx.x * 16;
  const int b  = blockIdx.y;
  const int t  = threadIdx.x;

  __shared__ int   idxs[16 * Kc];   // [n_local][k]
  __shared__ float fs[Cc * 17];     // feat tile [c][n] (pad 17)

  {
    int nl = t >> 4, k = t & 15;    // 256 threads == 16*16 indices
    idxs[t] = nbr[((size_t)b * Nc + n0 + nl) * Kc + k];
  }
  __syncthreads();

  const int c  = t & 63;
  const int ni = t >> 6;            // 0..3
#pragma unroll
  for (int p = 0; p < 4; ++p) {
    const int nl = ni * 4 + p;      // successive p -> contiguous x_info chunks
    const int n  = n0 + nl;
    const float qv = qT[((size_t)b * Nc + n) * Cc + c];
    const float* xp = x_info + (((size_t)b * Cc + c) * Nc + n) * Kc;
    float xw[16];
    *(float4*)(xw + 0)  = *(const float4*)(xp + 0);
    *(float4*)(xw + 4)  = *(const float4*)(xp + 4);
    *(float4*)(xw + 8)  = *(const float4*)(xp + 8);
    *(float4*)(xw + 12) = *(const float4*)(xp + 12);
    if (p < 3) __builtin_prefetch(xp + 16, 0, 0);   // global_prefetch_b8: next point

    float wv[16];
    float wmax = -3.402823466e+38f;
#pragma unroll
    for (int k = 0; k < Kc; ++k) {
      int id = idxs[nl * Kc + k];
      float kg = kT[((size_t)b * Nc + id) * Cc + c];  // coalesced 128B/wave gather
      float val = xw[k] * (kg - qv);
      wv[k] = val;
      wmax = fmaxf(wmax, val);
    }
    float esum = 0.f, acc = 0.f;
#pragma unroll
    for (int k = 0; k < Kc; ++k) {
      int id = idxs[nl * Kc + k];
      float vg = vT[((size_t)b * Nc + id) * Cc + c];
      float e = __expf(wv[k] - wmax);
      esum += e;
      acc += e * vg;
    }
    fs[c * 17 + nl] = acc / esum;
  }
  __syncthreads();

  // Final projection: out[o][n] = m2_w[128x64] @ feat[64x16], 8 waves x 1 tile.
  const int w = t >> 5, lane = t & 31, lrow = lane & 15, hi = lane >> 4;
  v8f acc = {0, 0, 0, 0, 0, 0, 0, 0};
  const float* Ap = m2_w + (16 * w + lrow) * Cc;
#pragma unroll
  for (int kk = 0; kk < 16; ++kk) {
    int k0 = 4 * kk + 2 * hi;
    v2f a;  a.x  = Ap[k0];              a.y  = Ap[k0 + 1];
    v2f bm; bm.x = fs[k0 * 17 + lrow];  bm.y = fs[(k0 + 1) * 17 + lrow];
    acc = wmma4(a, bm, acc);
  }
  float* ap = (float*)&acc;
#pragma unroll
  for (int r = 0; r < 8; ++r) {
    int o = 16 * w + r + 8 * hi;
    float sc = m2_g[o] * __frsqrt_rn(m2_v[o] + 1e-6f);
    float y = (ap[r] - m2_m[o]) * sc + m2_b[o];
    y = (y >= 0.f) ? y : 0.2f * y;
    out[((size_t)b * Oc + o) * Nc + n0 + lrow] = y;
  }
}

// ---------------------------------------------------------------------------
extern "C" void kernel_launch(void* const* d_in, const int* in_sizes, int n_in,
                              void* d_out, int out_size, void* d_ws,
                              size_t ws_size, hipStream_t stream) {
  (void)in_sizes; (void)n_in; (void)out_size; (void)ws_size;
  const float* x_info  = (const float*)d_in[0];
  const float* feature = (const float*)d_in[1];
  const int*   nbr     = (const int*)d_in[2];
  const float* Wq_w = (const float*)d_in[3];
  const float* Wq_g = (const float*)d_in[4];
  const float* Wq_b = (const float*)d_in[5];
  const float* Wq_m = (const float*)d_in[6];
  const float* Wq_v = (const float*)d_in[7];
  const float* Wk_w = (const float*)d_in[8];
  const float* Wk_g = (const float*)d_in[9];
  const float* Wk_b = (const float*)d_in[10];
  const float* Wk_m = (const float*)d_in[11];
  const float* Wk_v = (const float*)d_in[12];
  const float* Wv_w = (const float*)d_in[13];
  const float* Wv_g = (const float*)d_in[14];
  const float* Wv_b = (const float*)d_in[15];
  const float* Wv_m = (const float*)d_in[16];
  const float* Wv_v = (const float*)d_in[17];
  const float* m2_w = (const float*)d_in[18];
  const float* m2_g = (const float*)d_in[19];
  const float* m2_b = (const float*)d_in[20];
  const float* m2_m = (const float*)d_in[21];
  const float* m2_v = (const float*)d_in[22];
  float* out = (float*)d_out;

  const size_t plane = (size_t)Bc * Nc * Cc;  // floats per [B,N,C] tensor
  float* qT = (float*)d_ws;
  float* kT = qT + plane;
  float* vT = kT + plane;

  dim3 g(Nc / 16, Bc);
  qkv_kernel<<<g, dim3(128), 0, stream>>>(feature,
                                          Wq_w, Wq_g, Wq_b, Wq_m, Wq_v,
                                          Wk_w, Wk_g, Wk_b, Wk_m, Wk_v,
                                          Wv_w, Wv_g, Wv_b, Wv_m, Wv_v,
                                          qT, kT, vT);
  attn_kernel<<<g, dim3(256), 0, stream>>>(x_info, nbr, qT, kT, vT,
                                           m2_w, m2_g, m2_b, m2_m, m2_v, out);
}